// blendv2d_39453569581120
// MI455X (gfx1250) — compile-verified
//
#include <hip/hip_runtime.h>
#include <stdint.h>
#include <math.h>

#define T_STEPS 730
#define N_GRID  4000
#define NMUL    4
#define NZf     1e-5f
#define UH_LEN  15

__device__ __forceinline__ float sigmoidf(float x) { return 1.0f / (1.0f + expf(-x)); }

// argmax-first over 3 values (replicates jnp.argmax tie semantics)
__device__ __forceinline__ int sel3(float a, float b, float c) {
  int j = 0; float m = a;
  if (b > m) { j = 1; m = b; }
  if (c > m) { j = 2; }
  return j;
}

// ---------------- kernel 1: routing-parameter sigmoids + gamma unit hydrographs ----
__global__ __launch_bounds__(256) void uh_kernel(const float* __restrict__ hp,
                                                 float* __restrict__ uh1,
                                                 float* __restrict__ uh2) {
  int g = blockIdx.x * 256 + threadIdx.x;
  if (g >= N_GRID) return;
  const float* r = hp + (size_t)g * 100 + 96;
  float a1 = 0.30f + sigmoidf(r[0]) * (20.0f - 0.30f);
  float b1 = 0.01f + sigmoidf(r[1]) * (5.0f  - 0.01f);
  float a2 = 0.50f + sigmoidf(r[2]) * (13.0f - 0.50f);
  float b2 = 0.15f + sigmoidf(r[3]) * (1.5f  - 0.15f);
  float lg1 = lgammaf(a1), lg2 = lgammaf(a2);
  float lb1 = logf(b1),    lb2 = logf(b2);
  float w1[UH_LEN], w2[UH_LEN], s1 = 0.0f, s2 = 0.0f;
  #pragma unroll
  for (int k = 0; k < UH_LEN; ++k) {
    float t  = 0.5f + (float)k;
    float lt = logf(t);
    float v1 = expf((a1 - 1.0f) * lt - t / b1 - a1 * lb1 - lg1);
    float v2 = expf((a2 - 1.0f) * lt - t / b2 - a2 * lb2 - lg2);
    w1[k] = v1; s1 += v1;
    w2[k] = v2; s2 += v2;
  }
  float i1 = 1.0f / s1, i2 = 1.0f / s2;
  #pragma unroll
  for (int k = 0; k < UH_LEN; ++k) {
    uh1[(size_t)g * UH_LEN + k] = w1[k] * i1;
    uh2[(size_t)g * UH_LEN + k] = w2[k] * i2;
  }
}

// ---------------- kernel 2: sequential hydrological scan --------------------------
// One thread per (grid, nmul) chain.  The 420 MB weights stream (+ forcings) is
// double^2-buffered through LDS with per-wave-self-contained, fully coalesced
// global_load_async_to_lds_b128 batches, 4 timesteps deep (ASYNCcnt pipeline).
__global__ __launch_bounds__(256) void scan_kernel(const float* __restrict__ xp,
                                                   const float* __restrict__ wp,
                                                   const float* __restrict__ hp,
                                                   float* __restrict__ qq,
                                                   float* __restrict__ qs) {
  __shared__ float wsh[4][2304];   // 4 buffers x (64 grid cells x 36 weights)
  __shared__ float xsh[4][192];    // 4 buffers x (64 grid cells x 3 forcings)

  const int lt    = threadIdx.x;
  const int lane  = lt & 31;
  const int wvid  = lt >> 5;                 // wave id in block (0..7)
  const int tid   = blockIdx.x * 256 + lt;
  const int g     = tid >> 2;
  const int m     = tid & 3;
  const int gi    = (g < N_GRID) ? g : (N_GRID - 1);   // keep all lanes active
  const int gl    = lt >> 2;                 // grid cell local to block (0..63)
  const bool wr   = (m == 0) && (g < N_GRID);

  // --- async streaming setup (per-wave self-contained chunks) ---
  const uint64_t wbase  = (uint64_t)(uintptr_t)wp;
  const uint64_t xbase  = (uint64_t)(uintptr_t)xp;
  const uint32_t wstep  = (uint32_t)N_GRID * 144u;               // bytes / timestep
  const uint32_t xstep  = (uint32_t)N_GRID * 12u;
  const uint32_t wlimit = (uint32_t)T_STEPS * wstep - 16u;       // OOB clamp
  const uint32_t xlimit = (uint32_t)T_STEPS * xstep - 16u;
  uint32_t gw = (uint32_t)blockIdx.x * 64u * 144u;               // batch t=0 base
  uint32_t gx = (uint32_t)blockIdx.x * 64u * 12u;

  // wave w owns 1152 contiguous bytes (72 16B chunks) of each step's weights
  const uint32_t wc0 = (uint32_t)(72 * wvid + lane) * 16u;
  const uint32_t wc1 = wc0 + 512u;
  const uint32_t wc2 = (uint32_t)(72 * wvid + ((lane + 64 > 71) ? 71 : lane + 64)) * 16u;
  const uint32_t xc  = (uint32_t)(6 * wvid + ((lane > 5) ? 5 : lane)) * 16u;

  const uint32_t lwb = (uint32_t)(uintptr_t)&wsh[0][0];
  const uint32_t lxb = (uint32_t)(uintptr_t)&xsh[0][0];

  auto issue_batch = [&](uint32_t buf, uint32_t gwt, uint32_t gxt) {
    const uint32_t lb = lwb + buf * 9216u;
    const uint32_t xb = lxb + buf * 768u;
    uint32_t a;
    a = gwt + wc0; a = (a > wlimit) ? wlimit : a;
    asm volatile("global_load_async_to_lds_b128 %0, %1, %2"
                 :: "v"(lb + wc0), "v"(a), "s"(wbase) : "memory");
    a = gwt + wc1; a = (a > wlimit) ? wlimit : a;
    asm volatile("global_load_async_to_lds_b128 %0, %1, %2"
                 :: "v"(lb + wc1), "v"(a), "s"(wbase) : "memory");
    a = gwt + wc2; a = (a > wlimit) ? wlimit : a;
    asm volatile("global_load_async_to_lds_b128 %0, %1, %2"
                 :: "v"(lb + wc2), "v"(a), "s"(wbase) : "memory");
    a = gxt + xc;  a = (a > xlimit) ? xlimit : a;
    asm volatile("global_load_async_to_lds_b128 %0, %1, %2"
                 :: "v"(xb + xc), "v"(a), "s"(xbase) : "memory");
  };

  // prologue: 4 batches in flight (t = 0..3); params load below overlaps them
  issue_batch(0u, gw,              gx);
  issue_batch(1u, gw + wstep,      gx + xstep);
  issue_batch(2u, gw + 2u * wstep, gx + 2u * xstep);
  issue_batch(3u, gw + 3u * wstep, gx + 3u * xstep);

  // --- per-chain physical parameters (sigmoid + bounds), PHY_BOUNDS order ---
  const float* hb = hp + (size_t)gi * 100;
  #define LOADP(i, lo, hi) ((lo) + ((hi) - (lo)) * sigmoidf(hb[(i) * 4 + m]))
  const float ddf_min   = LOADP(0, 0.0f, 20.0f);
  const float ddf_max   = ddf_min + LOADP(1, 0.0f, 20.0f);
  const float Kcum      = LOADP(2, 0.01f, 0.2f);
  const float Kf_hbv    = LOADP(3, 0.0f, 5.0f);
  const float Kf_hmets  = LOADP(4, 0.0f, 5.0f);
  const float exp_fe    = LOADP(5, 0.0f, 1.0f);
  const float Tbf       = LOADP(6, -5.0f, 2.0f);
  const float Ccum      = LOADP(7, 0.005f, 0.05f);
  const float SWI       = LOADP(8, 0.0f, 0.4f);
  const float Tbm       = LOADP(9, -2.0f, 3.0f);
  const float fcmin     = LOADP(10, 0.0f, 0.1f);
  const float fc_sum    = fcmin + LOADP(11, 0.01f, 0.25f);
  const float vmax      = LOADP(12, 0.001f, 500.0f);
  const float alpha_hm  = LOADP(13, 0.0f, 1.0f);
  const float vic_beta  = LOADP(14, 0.1f, 3.0f);
  const float hbv_beta  = LOADP(15, 0.5f, 3.0f);
  const float qf_pow    = powf(10.0f, LOADP(16, -5.0f, -1.0f));
  const float qn        = LOADP(17, 0.5f, 2.0f);
  const float mmax      = LOADP(18, 0.0001f, 100.0f);
  const float sth       = LOADP(19, 0.0001f, 0.5f);
  const float bk        = powf(10.0f, LOADP(20, -5.0f, -1.0f));
  const float ET_eff    = LOADP(21, 0.0f, 3.0f);
  const float c_v2p     = LOADP(22, 1e-5f, 0.02f);
  const float c_phr     = LOADP(23, 1e-5f, 0.01f);
  #undef LOADP
  const float inv_vmax  = 1.0f / vmax;
  const float inv_1msth = 1.0f / (1.0f - sth);

  // --- state ---
  float sp0 = NZf, sp1 = NZf, sp2 = NZf;
  float lw0 = NZf, lw1 = NZf;
  float cm0 = 0.0f, cm1 = 0.0f, cm2 = 0.0f;
  float soil = NZf, phre = NZf;

  for (int t = 0; t < T_STEPS; ++t) {
    const uint32_t buf = (uint32_t)(t & 3);
    const int rem = T_STEPS - 1 - t;   // future batches still outstanding
    if (rem >= 3)      asm volatile("s_wait_asynccnt 12" ::: "memory");
    else if (rem == 2) asm volatile("s_wait_asynccnt 8"  ::: "memory");
    else if (rem == 1) asm volatile("s_wait_asynccnt 4"  ::: "memory");
    else               asm volatile("s_wait_asynccnt 0"  ::: "memory");

    const float* wvp = &wsh[buf][gl * 36 + m];
    const float w0 = wvp[0],  w1 = wvp[4],  w2 = wvp[8];
    const float w3 = wvp[12], w4 = wvp[16], w5 = wvp[20];
    const float w6 = wvp[24], w7 = wvp[28], w8 = wvp[32];
    const float* xvp = &xsh[buf][gl * 3];
    const float p  = xvp[0];
    const float tm = xvp[1];
    const float pe = xvp[2];

    // make sure our LDS reads retired before the async engine reuses this buffer
    asm volatile("s_wait_dscnt 0" ::: "memory");
    if (t + 4 < T_STEPS) issue_batch(buf, gw + 4u * wstep, gx + 4u * xstep);
    if (t + 8 < T_STEPS) {  // prime L2 ~8 steps ahead (global_prefetch_b8)
      __builtin_prefetch((const char*)wp + (gw + 8u * wstep + wc0), 0, 1);
      __builtin_prefetch((const char*)wp + (gw + 8u * wstep + wc1), 0, 1);
    }
    gw += wstep; gx += xstep;

    const float sf = (tm < 0.0f) ? p : 0.0f;
    const float rf = (tm < 0.0f) ? 0.0f : p;
    const float tfac = fmaxf(Tbf - tm, NZf);

    // --- snobal HBV ---
    float ovA;
    {
      float ddf  = fminf(ddf_max, ddf_min * (1.0f + Kcum * cm0));
      float pot  = fmaxf(ddf * (tm - Tbm), 0.0f);
      float refr = fminf(Kf_hbv * tfac, lw0);
      sp0 = sp0 + sf + refr;
      float melt = fminf(pot, sp0 + sf + refr);
      cm0 = (sp0 > NZf) ? (cm0 + melt) : 0.0f;
      sp0 = fmaxf(sp0 - melt, NZf);
      float wret = SWI * sp0;
      float wtmp = lw0 + melt + rf;
      ovA = fmaxf(wtmp - wret, 0.0f);
      lw0 = (ovA > 0.0f) ? wret : wtmp;
    }
    // --- snobal HMETS ---
    float ovB;
    {
      float ddf  = fminf(ddf_max, ddf_min * (1.0f + Kcum * cm1));
      float pot  = fmaxf(ddf * (tm - Tbm), 0.0f);
      float refr = fminf(Kf_hmets * powf(tfac, exp_fe), lw1);
      lw1 -= refr;
      sp1 += refr;
      float melt = fminf(pot, sp1 + sf);
      sp1 = sp1 + sf - melt;
      cm1 = (sp1 > NZf) ? (cm1 + melt) : 0.0f;
      float wrf  = fmaxf(fc_sum * (1.0f - Ccum * cm1), fcmin);
      float wret = wrf * sp1;
      float wtmp = lw1 + melt + rf;
      ovB = fmaxf(wtmp - wret, 0.0f);
      lw1 = (ovB > 0.0f) ? wret : wtmp;
      cm1 = (sp1 > NZf) ? (cm1 + melt) : 0.0f;   // reference updates cm twice
    }
    // --- snobal simple ---
    float ovC;
    {
      float ddf  = fminf(ddf_max, ddf_min * (1.0f + Kcum * cm2));
      float pot  = fmaxf(ddf * (tm - Tbm), 0.0f);
      float melt = fminf(pot, sp2);
      sp2 = sp2 + sf - melt;
      ovC = melt + rf;
      cm2 = (sp2 > NZf) ? (cm2 + melt) : 0.0f;
    }

    const int s0 = sel3(w0, w1, w2);
    const float ov = (s0 == 0) ? ovA : ((s0 == 1) ? ovB : ovC);

    // --- soil bucket ---
    const float sprop = fminf(fmaxf(soil * inv_vmax, 0.0f), 1.0f - NZf);
    const float omsp  = 1.0f - sprop;
    const float f_hbv   = fminf(fmaxf(powf(omsp, hbv_beta), 0.0f), 1.0f);
    const float f_hmets = fminf(fmaxf(1.0f - alpha_hm * sprop, 0.0f), 1.0f);
    const float f_vic   = fminf(fmaxf(1.0f - powf(omsp, vic_beta), 0.0f), 1.0f);
    const int s1 = sel3(w3, w4, w5);
    const float infil = ov * ((s1 == 0) ? f_hbv : ((s1 == 1) ? f_hmets : f_vic));
    float surface = ov - infil;
    const float et = fminf(pe * ET_eff * sprop, soil);
    const float q_exp = fminf(fmaxf(expf(-qf_pow * fmaxf(vmax - soil, NZf)), NZf), 1.0f);
    const float q_thr = powf(fminf(fmaxf((sprop - sth) * inv_1msth, NZf), 1.0f), qn);
    const float q_vic = fminf(fmaxf(powf(sprop, qn), NZf), 1.0f);
    const int s2 = sel3(w6, w7, w8);
    float quick = ((s2 == 0) ? q_exp : ((s2 == 1) ? q_thr : q_vic)) * mmax;
    quick = fminf(quick, fmaxf(soil - et, 0.0f));
    const float base = bk * soil;
    const float rech = c_v2p * soil;
    soil = soil + infil - et - quick - base - rech;
    const float excess = fmaxf(soil - vmax, 0.0f);
    soil = fminf(fmaxf(soil, NZf), vmax);
    surface += excess;
    const float base2 = c_phr * phre;
    phre = phre + rech - base2;

    // --- NMUL mean via 4-lane butterfly (wave32) ---
    float qqv = surface + quick;
    float qsv = base + base2;
    qqv += __shfl_xor(qqv, 1); qqv += __shfl_xor(qqv, 2);
    qsv += __shfl_xor(qsv, 1); qsv += __shfl_xor(qsv, 2);
    if (wr) {
      const size_t o = (size_t)t * N_GRID + g;
      qq[o] = 0.25f * qqv;
      qs[o] = 0.25f * qsv;
    }
  }
}

// ---------------- kernel 3: dual 15-tap causal FIR (gamma routing) ----------------
__global__ __launch_bounds__(256) void route_kernel(const float* __restrict__ qq,
                                                    const float* __restrict__ qs,
                                                    const float* __restrict__ uh1,
                                                    const float* __restrict__ uh2,
                                                    float* __restrict__ out) {
  const size_t tid = (size_t)blockIdx.x * 256 + threadIdx.x;
  if (tid >= (size_t)T_STEPS * N_GRID) return;
  const int t = (int)(tid / N_GRID);
  const int g = (int)(tid % N_GRID);
  const float* u1 = uh1 + (size_t)g * UH_LEN;
  const float* u2 = uh2 + (size_t)g * UH_LEN;
  const int kmax = (t < UH_LEN - 1) ? t : (UH_LEN - 1);
  float acc = 0.0f;
  for (int k = 0; k <= kmax; ++k) {
    const size_t o = (size_t)(t - k) * N_GRID + g;
    acc = fmaf(u1[k], qq[o], acc);
    acc = fmaf(u2[k], qs[o], acc);
  }
  out[tid] = acc;
}

extern "C" void kernel_launch(void* const* d_in, const int* in_sizes, int n_in,
                              void* d_out, int out_size, void* d_ws, size_t ws_size,
                              hipStream_t stream) {
  const float* xp = (const float*)d_in[0];   // x_phy        (730,4000,3)
  const float* wp = (const float*)d_in[1];   // weights      (730,4000,36)
  const float* hp = (const float*)d_in[2];   // hybrid_params(4000,100)
  float* out = (float*)d_out;

  float* ws  = (float*)d_ws;
  float* uh1 = ws;                                    // G*15
  float* uh2 = uh1 + (size_t)N_GRID * UH_LEN;         // G*15
  float* qq  = uh2 + (size_t)N_GRID * UH_LEN;         // T*G
  float* qs  = qq  + (size_t)T_STEPS * N_GRID;        // T*G

  uh_kernel<<<(N_GRID + 255) / 256, 256, 0, stream>>>(hp, uh1, uh2);

  const int nchain = ((N_GRID * NMUL) + 255) / 256;   // 63 blocks, all lanes active
  scan_kernel<<<nchain, 256, 0, stream>>>(xp, wp, hp, qq, qs);

  const size_t ntot = (size_t)T_STEPS * N_GRID;
  route_kernel<<<(unsigned)((ntot + 255) / 256), 256, 0, stream>>>(qq, qs, uh1, uh2, out);
}